// GCNConvBlock_17145509446019
// MI455X (gfx1250) — compile-verified
//
#include <hip/hip_runtime.h>

// Problem constants (match reference)
#define FIN  128
#define FOUT 128
#define EPSV 1e-5f

typedef float v2f __attribute__((ext_vector_type(2)));
typedef float v8f __attribute__((ext_vector_type(8)));

// ---------------------------------------------------------------------------
// Kernel 1: LayerNorm + ReLU.  One wave (32 lanes) per row; lane owns 4 floats.
// ---------------------------------------------------------------------------
__global__ __launch_bounds__(256) void ln_relu_kernel(
    const float* __restrict__ x, const float* __restrict__ gamma,
    const float* __restrict__ beta, float* __restrict__ xn, int n) {
  int wave = threadIdx.x >> 5;
  int lane = threadIdx.x & 31;
  int row  = blockIdx.x * 8 + wave;
  if (row >= n) return;

  float4 v = ((const float4*)(x + (size_t)row * FIN))[lane];
  float s  = v.x + v.y + v.z + v.w;
  float ss = v.x * v.x + v.y * v.y + v.z * v.z + v.w * v.w;
  // wave32 butterfly reduction
  for (int m = 16; m >= 1; m >>= 1) {
    s  += __shfl_xor(s,  m, 32);
    ss += __shfl_xor(ss, m, 32);
  }
  float mean = s * (1.0f / FIN);
  float var  = ss * (1.0f / FIN) - mean * mean;
  float rs   = rsqrtf(var + EPSV);

  float4 g  = ((const float4*)gamma)[lane];
  float4 bt = ((const float4*)beta)[lane];
  float4 o;
  o.x = fmaxf(0.0f, (v.x - mean) * rs * g.x + bt.x);
  o.y = fmaxf(0.0f, (v.y - mean) * rs * g.y + bt.y);
  o.z = fmaxf(0.0f, (v.z - mean) * rs * g.z + bt.z);
  o.w = fmaxf(0.0f, (v.w - mean) * rs * g.w + bt.w);
  ((float4*)(xn + (size_t)row * FIN))[lane] = o;
}

// ---------------------------------------------------------------------------
// Kernel 2: h = xn @ W^T  via V_WMMA_F32_16X16X4_F32 (fp32-exact matrix core).
// Block = 256 threads = 8 waves. Block covers 16 rows x all 128 output cols;
// wave w computes the 16x16 tile at cols [16w, 16w+16).
// A slab (16x128) staged in LDS, rows padded to 132 floats -> the per-iter
// 16x4 fragment ds_load_b64s touch all 64 banks exactly once (conflict-free).
// Fragment layouts per CDNA5 ISA 7.12.2:
//   A 16x4 f32 : lanes 0-15 hold {K=0,K=1}, lanes 16-31 hold {K=2,K=3}, M=lane%16
//   B 4x16 f32 : lanes 0-15 hold rows K=0,1 at N=lane; lanes 16-31 rows K=2,3
//   C/D 16x16  : VGPR v -> M=v (lanes 0-15) / M=v+8 (lanes 16-31), N=lane%16
// ---------------------------------------------------------------------------
#define A_PITCH 132

__global__ __launch_bounds__(256) void gemm_wmma_kernel(
    const float* __restrict__ xn, const float* __restrict__ W,
    float* __restrict__ h) {
  __shared__ float alds[16 * A_PITCH];
  const int tid = threadIdx.x;
  const int n0  = blockIdx.x * 16;

  // Stage A: 16 rows x 128 floats, coalesced float4 loads (pitch 528B = 16B-aligned)
  for (int s = tid; s < 16 * (FIN / 4); s += 256) {
    int r = s >> 5, c4 = s & 31;
    float4 v = ((const float4*)(xn + (size_t)(n0 + r) * FIN))[c4];
    *((float4*)&alds[r * A_PITCH + c4 * 4]) = v;
  }
  __syncthreads();

  const int wave  = tid >> 5;
  const int lane  = tid & 31;
  const int o0    = wave * 16;
  const int mrow  = lane & 15;        // M (for A) / N (for B,D)
  const int khalf = (lane >> 4) * 2;  // lanes 16-31 take K=2,3

  const float* abase = &alds[mrow * A_PITCH + khalf];
  const float* wbase = W + (size_t)(o0 + mrow) * FIN + khalf;  // B[k][n] = W[n][k]

  v8f c = {};
#pragma unroll 8
  for (int k = 0; k < FIN; k += 4) {
    v2f a = *((const v2f*)(abase + k));
    v2f b = *((const v2f*)(wbase + k));
    // 8 args: (neg_a, A, neg_b, B, c_mod, C, reuse_a, reuse_b)
    c = __builtin_amdgcn_wmma_f32_16x16x4_f32(
        false, a, false, b, (short)0, c, false, false);
  }

  const int col   = o0 + mrow;
  const int rbase = n0 + ((lane >> 4) << 3);  // +8 rows for upper half-wave
#pragma unroll
  for (int v = 0; v < 8; ++v) {
    h[(size_t)(rbase + v) * FOUT + col] = c[v];
  }
}

// ---------------------------------------------------------------------------
// Kernel 3a/3b: degree = 1.0 (self loop) + scatter-add of edge weights
// ---------------------------------------------------------------------------
__global__ void deg_init_kernel(float* __restrict__ deg, int n) {
  int i = blockIdx.x * 256 + threadIdx.x;
  if (i < n) deg[i] = 1.0f;
}

__global__ void deg_edge_kernel(const int* __restrict__ ei,
                                const float* __restrict__ ew,
                                float* __restrict__ deg, int e) {
  int i = blockIdx.x * 256 + threadIdx.x;
  if (i < e) unsafeAtomicAdd(&deg[ei[e + i]], ew[i]);  // col = ei[1*E + i]
}

// ---------------------------------------------------------------------------
// Kernel 4a: dinv = rsqrt(deg);  4b: out = b + h * dinv^2  (self-loop term,
// doubles as the zero-init of the poisoned d_out)
// ---------------------------------------------------------------------------
__global__ void dinv_kernel(const float* __restrict__ deg,
                            float* __restrict__ dinv, int n) {
  int i = blockIdx.x * 256 + threadIdx.x;
  if (i < n) {
    float d = deg[i];
    dinv[i] = (d > 0.0f) ? rsqrtf(d) : 0.0f;
  }
}

__global__ __launch_bounds__(128) void out_init_kernel(
    const float* __restrict__ h, const float* __restrict__ dinv,
    const float* __restrict__ b, float* __restrict__ out) {
  int i = blockIdx.x;          // node
  int f = threadIdx.x;         // feature, blockDim = 128
  float dv = dinv[i];
  size_t idx = (size_t)i * FOUT + f;
  out[idx] = b[f] + h[idx] * dv * dv;
}

// ---------------------------------------------------------------------------
// Kernel 5: edge aggregation. One wave per edge (wave32); lane owns a float4
// of the 128-feature message. out/h live in the 192MB L2 -> L2 atomics.
// ---------------------------------------------------------------------------
__global__ __launch_bounds__(256) void edge_agg_kernel(
    const int* __restrict__ ei, const float* __restrict__ ew,
    const float* __restrict__ h, const float* __restrict__ dinv,
    float* __restrict__ out, int e) {
  int eid  = blockIdx.x * 8 + (threadIdx.x >> 5);
  if (eid >= e) return;
  int lane = threadIdx.x & 31;
  int r = ei[eid];       // source
  int c = ei[e + eid];   // destination
  float norm = dinv[r] * ew[eid] * dinv[c];
  float4 hv = ((const float4*)(h + (size_t)r * FOUT))[lane];
  float* ob = out + (size_t)c * FOUT + lane * 4;
  unsafeAtomicAdd(ob + 0, hv.x * norm);
  unsafeAtomicAdd(ob + 1, hv.y * norm);
  unsafeAtomicAdd(ob + 2, hv.z * norm);
  unsafeAtomicAdd(ob + 3, hv.w * norm);
}

// ---------------------------------------------------------------------------
extern "C" void kernel_launch(void* const* d_in, const int* in_sizes, int n_in,
                              void* d_out, int out_size, void* d_ws, size_t ws_size,
                              hipStream_t stream) {
  const float* x     = (const float*)d_in[0];
  const int*   ei    = (const int*)d_in[1];   // [2, E] flat
  const float* ew    = (const float*)d_in[2];
  const float* gamma = (const float*)d_in[3];
  const float* beta  = (const float*)d_in[4];
  const float* W     = (const float*)d_in[5];
  const float* b     = (const float*)d_in[6];
  float* out = (float*)d_out;

  const int n = in_sizes[0] / FIN;   // 100000 (multiple of 16)
  const int e = in_sizes[2];         // 1600000

  // Workspace layout (floats): xn | h | deg | dinv  (~103 MB)
  float* xn   = (float*)d_ws;
  float* h    = xn + (size_t)n * FIN;
  float* deg  = h + (size_t)n * FOUT;
  float* dinv = deg + n;

  // 1) LayerNorm + ReLU
  ln_relu_kernel<<<(n + 7) / 8, 256, 0, stream>>>(x, gamma, beta, xn, n);

  // 2) h = xn @ W^T via fp32 WMMA (n is a multiple of 16)
  gemm_wmma_kernel<<<n / 16, 256, 0, stream>>>(xn, W, h);

  // 3) degrees
  deg_init_kernel<<<(n + 255) / 256, 256, 0, stream>>>(deg, n);
  deg_edge_kernel<<<(e + 255) / 256, 256, 0, stream>>>(ei, ew, deg, e);

  // 4) dinv + self-loop init of out
  dinv_kernel<<<(n + 255) / 256, 256, 0, stream>>>(deg, dinv, n);
  out_init_kernel<<<n, 128, 0, stream>>>(h, dinv, b, out);

  // 5) edge scatter-add
  edge_agg_kernel<<<(e + 7) / 8, 256, 0, stream>>>(ei, ew, h, dinv, out, e);

  (void)n_in; (void)out_size; (void)ws_size;
}